// OTStyleModel_75488345194548
// MI455X (gfx1250) — compile-verified
//
#include <hip/hip_runtime.h>
#include <hip/hip_bf16.h>
#include <math.h>

#define LSEQ 128
#define HDIM 768
#define CDIM 192
#define SDIM 128
#define NFEA 321          // CD + SD + 1
#define EPSV 0.05f
#define INV_EPS 20.0f
#define NEGC -1000000000.0f
#define NITER 50
#define CP 129            // padded C row (conflict-free row & col sweeps)
#define PP 36             // padded K-panel row: 16B-aligned rows AND conflict-free columns
#define NPAN (HDIM / 32)  // 24 K-panels

typedef float v2f __attribute__((ext_vector_type(2)));
typedef float v8f __attribute__((ext_vector_type(8)));

// Issue one 128x32 f32 panel (rows pitch PP) as 4 async b128 copies per thread.
// ASYNCcnt += 4 per wave; completion is in-order, so s_wait_asynccnt<=4 after
// issuing the NEXT panel guarantees THIS panel has landed.
__device__ __forceinline__ void async_panel_issue(const float* __restrict__ Xkp,
                                                  unsigned lds_base, int tid) {
  #pragma unroll
  for (int t = 0; t < 4; ++t) {
    int f  = tid + t * 256;          // 0..1023 float4 tiles
    int r  = f >> 3;                 // row 0..127
    int c4 = f & 7;                  // float4 column 0..7
    unsigned loff = lds_base + (unsigned)((r * PP + c4 * 4) * 4);   // 16B aligned
    unsigned voff = (unsigned)((r * HDIM + c4 * 4) * 4);            // 16B aligned
    asm volatile("global_load_async_to_lds_b128 %0, %1, %2"
                 :: "v"(loff), "v"(voff), "s"(Xkp) : "memory");
  }
}

__global__ __launch_bounds__(256) void ot_sinkhorn_head_kernel(
    const float* __restrict__ Xall, const int* __restrict__ tt,
    const int* __restrict__ am,
    const float* __restrict__ Wc,  const float* __restrict__ bc,
    const float* __restrict__ Wsm, const float* __restrict__ bs,
    const float* __restrict__ gate,
    const float* __restrict__ g_ln, const float* __restrict__ b_ln,
    const float* __restrict__ Wcls, const float* __restrict__ bcls,
    float* __restrict__ outp)
{
  __shared__ float Cmat[LSEQ * CP];          // 66,048 B : Gram -> distances
  __shared__ float panelA[LSEQ * PP];        // 18,432 B : async dbl-buffer
  __shared__ float panelB[LSEQ * PP];        // 18,432 B
  __shared__ float repv[2 * HDIM];           //  6,144 B : [cls | mmean0-mmean1]
  __shared__ float feats[NFEA + 3];
  __shared__ float fm0[LSEQ], fm1[LSEQ], loga[LSEQ], logb[LSEQ], av[LSEQ], bv[LSEQ];
  __shared__ float fv[LSEQ], gv[LSEQ], p0v[LSEQ], p1v[LSEQ], w0v[LSEQ], w1v[LSEQ];
  __shared__ float pmax[256], psum[256], sqv[LSEQ];
  __shared__ float sc[4];                    // n0, n1, div

  const int tid  = threadIdx.x;
  const int b    = blockIdx.x;
  const int lane = tid & 31;
  const int wv   = tid >> 5;                 // 8 wave32 waves
  const float* X = Xall + (size_t)b * LSEQ * HDIM;

  const unsigned ldsA = (unsigned)(size_t)(void*)panelA;   // LDS byte offsets
  const unsigned ldsB = (unsigned)(size_t)(void*)panelB;

  // ---------------- phase 0: masks / Sinkhorn weights ----------------
  if (tid < LSEQ) {
    int t = tt[b * LSEQ + tid], a = am[b * LSEQ + tid];
    fm0[tid] = (a == 1 && t == 0) ? 1.f : 0.f;
    fm1[tid] = (a == 1 && t == 1) ? 1.f : 0.f;
    fv[tid] = 0.f; gv[tid] = 0.f; p0v[tid] = 0.f; p1v[tid] = 0.f;
  }
  __syncthreads();
  if (tid == 0) {
    float c0 = 0.f, c1 = 0.f;
    for (int l = 0; l < LSEQ; ++l) { c0 += fm0[l]; c1 += fm1[l]; }
    sc[0] = fmaxf(c0, 1.f); sc[1] = fmaxf(c1, 1.f);
  }
  __syncthreads();
  const float n0 = sc[0], n1 = sc[1];
  if (tid < LSEQ) {
    loga[tid] = fm0[tid] > 0.f ? -__logf(n0) : NEGC;
    logb[tid] = fm1[tid] > 0.f ? -__logf(n1) : NEGC;
    av[tid]   = fm0[tid] > 0.f ? 1.f / n0 : 0.f;
    bv[tid]   = fm1[tid] > 0.f ? 1.f / n1 : 0.f;
  }

  // ---------------- phase 1: Gram G = X*X^T via f32 WMMA, async dbl-buffered ----------------
  v8f acc[8];
  #pragma unroll
  for (int i = 0; i < 8; ++i)
    #pragma unroll
    for (int j = 0; j < 8; ++j) acc[i][j] = 0.f;

  const int lrow  = lane & 15;
  const int lhalf = lane >> 4;
  const int mbase = wv * 16;
  const int cN    = tid & 31;      // rep-accum column within panel
  const int grp   = tid >> 5;      // rep-accum row group (16 rows)

  async_panel_issue(X, ldsA, tid);               // prologue: panel 0 -> A
  for (int p = 0; p < NPAN; ++p) {
    const float* pan = (p & 1) ? panelB : panelA;
    const int kp = p * 32;
    if (p + 1 < NPAN) {
      async_panel_issue(X + (size_t)(kp + 32), (p & 1) ? ldsA : ldsB, tid);
      asm volatile("s_wait_asynccnt 0x4" ::: "memory");   // panel p landed
    } else {
      asm volatile("s_wait_asynccnt 0x0" ::: "memory");
    }
    __syncthreads();

    // --- WMMA over this K-panel (uniform: EXEC all ones) ---
    #pragma unroll
    for (int kk = 0; kk < 32; kk += 4) {
      v2f afr;  // A frag 16x4 f32: lane<16 K=kk..kk+1, lane>=16 K=kk+2..kk+3
      afr.x = pan[(mbase + lrow) * PP + kk + 2 * lhalf];
      afr.y = pan[(mbase + lrow) * PP + kk + 2 * lhalf + 1];
      #pragma unroll
      for (int nt = 0; nt < 8; ++nt) {
        v2f bfr; // B frag of X^T: same loader with row base nt*16
        bfr.x = pan[(nt * 16 + lrow) * PP + kk + 2 * lhalf];
        bfr.y = pan[(nt * 16 + lrow) * PP + kk + 2 * lhalf + 1];
        acc[nt] = __builtin_amdgcn_wmma_f32_16x16x4_f32(
            false, afr, false, bfr, (short)0, acc[nt], false, false);
      }
    }

    // --- fold masked column sums (mmean) into this pass: X read from HBM once ---
    {
      float s0 = 0.f, s1 = 0.f;
      #pragma unroll 4
      for (int rr = 0; rr < 16; ++rr) {
        int row = grp * 16 + rr;
        float v = pan[row * PP + cN];   // lanes -> consecutive addrs: conflict-free
        s0 += fm0[row] * v; s1 += fm1[row] * v;
      }
      pmax[tid] = s0; psum[tid] = s1;
    }
    __syncthreads();
    if (tid < 32) {
      float s0 = 0.f, s1 = 0.f;
      #pragma unroll
      for (int g = 0; g < 8; ++g) { s0 += pmax[g * 32 + tid]; s1 += psum[g * 32 + tid]; }
      repv[kp + tid]        = pan[tid];            // cls = out[b][0][:]
      repv[HDIM + kp + tid] = s0 / n0 - s1 / n1;   // mmean0 - mmean1
    }
    __syncthreads();   // panel consumed; safe to overwrite next round
  }

  // C/D layout: VGPR r -> M = mbase + r + 8*lhalf, N = nt*16 + (lane&15)
  #pragma unroll
  for (int nt = 0; nt < 8; ++nt)
    #pragma unroll
    for (int r8 = 0; r8 < 8; ++r8)
      Cmat[(mbase + r8 + 8 * lhalf) * CP + nt * 16 + lrow] = acc[nt][r8];
  __syncthreads();

  // squared norms = Gram diagonal; C = sqrt(clip(sq_i + sq_j - 2G, 1e-12))
  if (tid < LSEQ) sqv[tid] = Cmat[tid * CP + tid];
  __syncthreads();
  for (int e = tid; e < LSEQ * LSEQ; e += 256) {
    int i = e >> 7, j = e & 127;
    float d2 = sqv[i] + sqv[j] - 2.f * Cmat[i * CP + j];
    Cmat[i * CP + j] = sqrtf(fmaxf(d2, 1e-12f));
  }
  __syncthreads();

  // ---------------- phase 2: Sinkhorn cross iterations (2 threads/row) ----------------
  const int r    = tid & 127;
  const int half = tid >> 7;
  const int j0   = half * 64;
  for (int it = 0; it < NITER; ++it) {
    if (tid < LSEQ) w0v[tid] = EPSV * logb[tid] + gv[tid];
    __syncthreads();
    float m = -3.4e38f;
    for (int j = j0; j < j0 + 64; ++j) m = fmaxf(m, w0v[j] - Cmat[r * CP + j]);
    pmax[tid] = m; __syncthreads();
    float M = fmaxf(pmax[r], pmax[r + 128]);
    float s = 0.f;
    for (int j = j0; j < j0 + 64; ++j)
      s += __expf((w0v[j] - Cmat[r * CP + j] - M) * INV_EPS);
    psum[tid] = s; __syncthreads();
    if (tid < LSEQ) fv[tid] = -M - EPSV * __logf(psum[tid] + psum[tid + 128]);
    __syncthreads();
    if (tid < LSEQ) w1v[tid] = EPSV * loga[tid] + fv[tid];
    __syncthreads();
    m = -3.4e38f;
    for (int i = j0; i < j0 + 64; ++i) m = fmaxf(m, w1v[i] - Cmat[i * CP + r]);
    pmax[tid] = m; __syncthreads();
    M = fmaxf(pmax[r], pmax[r + 128]);
    s = 0.f;
    for (int i = j0; i < j0 + 64; ++i)
      s += __expf((w1v[i] - Cmat[i * CP + r] - M) * INV_EPS);
    psum[tid] = s; __syncthreads();
    if (tid < LSEQ) gv[tid] = -M - EPSV * __logf(psum[tid] + psum[tid + 128]);
    __syncthreads();
  }

  // ---------------- phase 3: both symmetric scans simultaneously ----------------
  for (int it = 0; it < NITER; ++it) {
    if (tid < LSEQ) w0v[tid] = EPSV * loga[tid] + p0v[tid];
    else            w1v[tid - 128] = EPSV * logb[tid - 128] + p1v[tid - 128];
    __syncthreads();
    const float* wa = (tid < LSEQ) ? w0v : w1v;
    float pr = (tid < LSEQ) ? p0v[r] : p1v[r];
    float m = -3.4e38f;
    for (int j = 0; j < LSEQ; ++j) m = fmaxf(m, wa[j] - Cmat[r * CP + j]);
    float s = 0.f;
    for (int j = 0; j < LSEQ; ++j)
      s += __expf((wa[j] - Cmat[r * CP + j] - m) * INV_EPS);
    float np = 0.5f * (pr - m - EPSV * __logf(s));
    __syncthreads();
    if (tid < LSEQ) p0v[r] = np; else p1v[r] = np;
    __syncthreads();
  }

  if (tid == 0) {
    float d = 0.f;
    for (int l = 0; l < LSEQ; ++l)
      d += av[l] * (fv[l] - p0v[l]) + bv[l] * (gv[l] - p1v[l]);
    sc[2] = d;
  }
  __syncthreads();

  // ---------------- phase 4: head (rep already built in phase 1) ----------------
  const float gsig = 1.f / (1.f + __expf(-gate[0]));
  for (int f = wv; f < CDIM + SDIM; f += 8) {
    const float* wr; float bias, scale;
    if (f < CDIM) { wr = Wc  + (size_t)f * 2 * HDIM;          bias = bc[f];        scale = 1.f - gsig; }
    else          { wr = Wsm + (size_t)(f - CDIM) * 2 * HDIM; bias = bs[f - CDIM]; scale = gsig; }
    float s = 0.f;
    for (int k = lane; k < 2 * HDIM; k += 32) s += repv[k] * wr[k];
    #pragma unroll
    for (int off = 16; off; off >>= 1) s += __shfl_xor(s, off, 32);
    if (lane == 0) feats[f] = (s + bias) * scale;
  }
  __syncthreads();
  if (tid == 0) feats[CDIM + SDIM] = sc[2];
  __syncthreads();
  // layernorm over 321 feats
  float ls = 0.f, lq = 0.f;
  for (int f = tid; f < NFEA; f += 256) { float v = feats[f]; ls += v; lq += v * v; }
  pmax[tid] = ls; psum[tid] = lq; __syncthreads();
  for (int st = 128; st; st >>= 1) {
    if (tid < st) { pmax[tid] += pmax[tid + st]; psum[tid] += psum[tid + st]; }
    __syncthreads();
  }
  const float mu   = pmax[0] / (float)NFEA;
  const float var  = psum[0] / (float)NFEA - mu * mu;
  const float rstd = rsqrtf(var + 1e-5f);
  if (tid < 32) {
    float s0 = 0.f, s1 = 0.f;
    for (int f = tid; f < NFEA; f += 32) {
      float fn = (feats[f] - mu) * rstd * g_ln[f] + b_ln[f];
      s0 += fn * Wcls[f]; s1 += fn * Wcls[NFEA + f];
    }
    #pragma unroll
    for (int off = 16; off; off >>= 1) {
      s0 += __shfl_xor(s0, off, 32); s1 += __shfl_xor(s1, off, 32);
    }
    if (tid == 0) {
      outp[b * 2 + 0] = s0 + bcls[0];
      outp[b * 2 + 1] = s1 + bcls[1];
    }
  }
}

// ---------------- ortho = mean((Wc @ Ws^T)^2), deterministic 2-stage ----------------
__global__ __launch_bounds__(256) void ortho_partial_kernel(
    const float* __restrict__ Wc, const float* __restrict__ Wsm,
    float* __restrict__ part)
{
  __shared__ float red[256];
  int bx = blockIdx.x;                 // 96 tiles: 12 x 8 of 16x16
  int rr = (bx >> 3) * 16 + (threadIdx.x >> 4);
  int cc = (bx & 7)  * 16 + (threadIdx.x & 15);
  const float* a = Wc  + (size_t)rr * 2 * HDIM;
  const float* b = Wsm + (size_t)cc * 2 * HDIM;
  float s = 0.f;
  for (int k = 0; k < 2 * HDIM; ++k) s += a[k] * b[k];
  red[threadIdx.x] = s * s;
  __syncthreads();
  for (int st = 128; st; st >>= 1) {
    if (threadIdx.x < st) red[threadIdx.x] += red[threadIdx.x + st];
    __syncthreads();
  }
  if (threadIdx.x == 0) part[bx] = red[0];
}

__global__ void ortho_final_kernel(const float* __restrict__ part, float* __restrict__ outp)
{
  if (threadIdx.x == 0) {
    float s = 0.f;
    for (int i = 0; i < 96; ++i) s += part[i];
    outp[512] = s / (float)(CDIM * SDIM);
  }
}

extern "C" void kernel_launch(void* const* d_in, const int* in_sizes, int n_in,
                              void* d_out, int out_size, void* d_ws, size_t ws_size,
                              hipStream_t stream) {
  const float* out_t = (const float*)d_in[0];
  const int*   tt    = (const int*)d_in[1];
  const int*   am    = (const int*)d_in[2];
  const float* Wc    = (const float*)d_in[3];
  const float* bc    = (const float*)d_in[4];
  const float* Wsm   = (const float*)d_in[5];
  const float* bs    = (const float*)d_in[6];
  const float* gate  = (const float*)d_in[7];
  const float* g_ln  = (const float*)d_in[8];
  const float* b_ln  = (const float*)d_in[9];
  const float* Wcls  = (const float*)d_in[10];
  const float* bcls  = (const float*)d_in[11];
  float* outp = (float*)d_out;
  float* ws   = (float*)d_ws;

  ot_sinkhorn_head_kernel<<<256, 256, 0, stream>>>(
      out_t, tt, am, Wc, bc, Wsm, bs, gate, g_ln, b_ln, Wcls, bcls, outp);
  ortho_partial_kernel<<<96, 256, 0, stream>>>(Wc, Wsm, ws);
  ortho_final_kernel<<<1, 32, 0, stream>>>(ws, outp);
}